// RocketLeagueGCN_18030272709301
// MI455X (gfx1250) — compile-verified
//
#include <hip/hip_runtime.h>
#include <hip/hip_bf16.h>

// ---------------------------------------------------------------------------
// RocketLeagueGCN on MI455X (gfx1250, wave32)
//
//  1. degree + rsqrt                      (edge atomics, trivial)
//  2. GEMM  xw = x  @ W1   [131072x32 * 32x64]  -> v_wmma_f32_16x16x4_f32
//     (one wave per 16x64 strip: 4 independent accumulators share one A frag)
//  3. edge scatter-add (normalized)       (memory/atomic bound, L2-resident)
//  4. fused self-loop + bias + BN stats, then BN+ReLU
//  5. layer 2 (same shape, K=64)
//  6. graph mean-pool (atomics), 2 MLP heads (one wave32 per graph)
// ---------------------------------------------------------------------------

#define NNODES 131072
#define NEDGES 2097152
#define FIN    32
#define HID    64
#define GFEAT  16
#define NGRAPH 1024
#define BN_EPS 1e-5f

typedef __attribute__((ext_vector_type(2))) float v2f;
typedef __attribute__((ext_vector_type(8))) float v8f;

// ---------------------------------------------------------------- fill -----
__global__ void k_fill(float* __restrict__ p, int n, float v) {
    int i = blockIdx.x * blockDim.x + threadIdx.x;
    if (i < n) p[i] = v;
}

// ------------------------------------------------------------- degree ------
__global__ void k_degree(const int* __restrict__ dst, float* __restrict__ deg) {
    int e = blockIdx.x * blockDim.x + threadIdx.x;
    if (e < NEDGES) atomicAdd(&deg[dst[e]], 1.0f);
}

__global__ void k_rsqrt_inplace(float* __restrict__ d) {
    int i = blockIdx.x * blockDim.x + threadIdx.x;
    if (i < NNODES) d[i] = rsqrtf(d[i]);
}

// ------------------------------------------- WMMA fp32 GEMM (Ncols = 64) ---
// One wave per 16-row x 64-col strip. Per K-step of 4: one A fragment feeds
// four V_WMMA_F32_16X16X4_F32 (independent accumulators -> no RAW hazards,
// 4x less A traffic than tile-per-wave).
// fp32 A layout (ISA 7.12.2): lanes 0-15: M=0..15/K={k0,k0+1}; lanes 16-31:
// K={k0+2,k0+3}. B mirrors with N in place of M. C/D: VGPR r -> row r
// (lanes 0-15) / row r+8 (lanes 16-31), col = lane&15.
__global__ void k_gemm_wmma(const float* __restrict__ A,   // [M x K] row major
                            const float* __restrict__ B,   // [K x 64] row major
                            float* __restrict__ C,         // [M x 64] row major
                            int M, int K) {
    const int N = HID;                      // 64
    __shared__ float Bs[HID * HID];         // <= 16 KB, whole weight matrix
    for (int i = threadIdx.x; i < K * N; i += blockDim.x) Bs[i] = B[i];
    __syncthreads();

    int wave = blockIdx.x * (blockDim.x >> 5) + (threadIdx.x >> 5);
    int lane = threadIdx.x & 31;
    int m0   = wave * 16;
    if (m0 >= M) return;                    // wave-uniform; EXEC stays all-1s
    int half = lane >> 4;                   // 0: K pair {0,1}, 1: {2,3}
    int l    = lane & 15;

    const float* arow = A + (size_t)(m0 + l) * K;
    v8f acc0 = {}, acc1 = {}, acc2 = {}, acc3 = {};
    for (int k0 = 0; k0 < K; k0 += 4) {
        int ka = k0 + half * 2;
        v2f a;
        a.x = arow[ka + 0];
        a.y = arow[ka + 1];
        __builtin_prefetch(arow + ka + 4, 0, 3);   // global_prefetch_b8 (near)
        const float* brow = &Bs[ka * N + l];
        v2f b0, b1, b2, b3;
        b0.x = brow[0];  b0.y = brow[N + 0];
        b1.x = brow[16]; b1.y = brow[N + 16];
        b2.x = brow[32]; b2.y = brow[N + 32];
        b3.x = brow[48]; b3.y = brow[N + 48];
        acc0 = __builtin_amdgcn_wmma_f32_16x16x4_f32(false, a, false, b0,
                                                     (short)0, acc0, false, false);
        acc1 = __builtin_amdgcn_wmma_f32_16x16x4_f32(false, a, false, b1,
                                                     (short)0, acc1, false, false);
        acc2 = __builtin_amdgcn_wmma_f32_16x16x4_f32(false, a, false, b2,
                                                     (short)0, acc2, false, false);
        acc3 = __builtin_amdgcn_wmma_f32_16x16x4_f32(false, a, false, b3,
                                                     (short)0, acc3, false, false);
    }

    float* crow = C + (size_t)(m0 + half * 8) * N + l;
#pragma unroll
    for (int r = 0; r < 8; ++r) {
        crow[(size_t)r * N + 0]  = acc0[r];
        crow[(size_t)r * N + 16] = acc1[r];
        crow[(size_t)r * N + 32] = acc2[r];
        crow[(size_t)r * N + 48] = acc3[r];
    }
}

// ---------------------------------------------- normalized edge scatter ----
// agg[dst, f] += xw[src, f] * dinv[src] * dinv[dst]   (one thread per (e,f))
__global__ void k_edge_scatter(const float* __restrict__ xw,
                               const int* __restrict__ src,
                               const int* __restrict__ dst,
                               const float* __restrict__ dinv,
                               float* __restrict__ agg) {
    int idx = blockIdx.x * blockDim.x + threadIdx.x;     // < NEDGES*64
    int e = idx >> 6, f = idx & 63;
    if (e >= NEDGES) return;
    int s = src[e], d = dst[e];
    float nrm = dinv[s] * dinv[d];
    atomicAdd(&agg[(size_t)d * HID + f], xw[(size_t)s * HID + f] * nrm);
}

// ------------------------- fused self-loop + bias + BN statistics ----------
// h = agg + xw * dinv^2 + bias   (written back), with per-feature block
// partial sums/sumsq reduced in LDS then 64-wide atomics.
__global__ void k_selfloop_bias_stats(float* __restrict__ agg,
                                      const float* __restrict__ xw,
                                      const float* __restrict__ dinv,
                                      const float* __restrict__ bias,
                                      float* __restrict__ sum,
                                      float* __restrict__ sumsq) {
    __shared__ float ssum[256], ssq[256];
    int f   = threadIdx.x & 63;
    int sub = threadIdx.x >> 6;              // 0..3
    float bf = bias[f];
    float ls = 0.f, lq = 0.f;
    for (int i = blockIdx.x * 4 + sub; i < NNODES; i += gridDim.x * 4) {
        float di = dinv[i];
        size_t idx = (size_t)i * HID + f;
        float v = agg[idx] + xw[idx] * di * di + bf;
        agg[idx] = v;
        ls += v; lq += v * v;
    }
    ssum[threadIdx.x] = ls; ssq[threadIdx.x] = lq;
    __syncthreads();
    if (sub == 0) {
        ls = ssum[f] + ssum[64 + f] + ssum[128 + f] + ssum[192 + f];
        lq = ssq[f]  + ssq[64 + f]  + ssq[128 + f]  + ssq[192 + f];
        atomicAdd(&sum[f], ls);
        atomicAdd(&sumsq[f], lq);
    }
}

// ---------------------------------------------- BN normalize + ReLU --------
__global__ void k_bn_relu(float* __restrict__ h,
                          const float* __restrict__ sum,
                          const float* __restrict__ sumsq,
                          const float* __restrict__ gamma,
                          const float* __restrict__ beta) {
    int idx = blockIdx.x * blockDim.x + threadIdx.x;     // < NNODES*64
    int f = idx & 63;
    const float invN = 1.0f / (float)NNODES;
    float mu  = sum[f] * invN;
    float var = sumsq[f] * invN - mu * mu;               // biased, matches ref
    float v = (h[idx] - mu) * rsqrtf(var + BN_EPS) * gamma[f] + beta[f];
    h[idx] = fmaxf(v, 0.0f);
}

// ---------------------------------------------- graph pooling --------------
__global__ void k_pool_sum(const float* __restrict__ h,
                           const int* __restrict__ batch,
                           float* __restrict__ pooled) {
    int idx = blockIdx.x * blockDim.x + threadIdx.x;     // < NNODES*64
    int i = idx >> 6, f = idx & 63;
    atomicAdd(&pooled[(size_t)batch[i] * HID + f], h[idx]);
}

__global__ void k_pool_cnt(const int* __restrict__ batch, float* __restrict__ cnt) {
    int i = blockIdx.x * blockDim.x + threadIdx.x;
    if (i < NNODES) atomicAdd(&cnt[batch[i]], 1.0f);
}

// ---------------------------------------------- MLP heads ------------------
// One wave32 per graph. lane j owns hidden unit j of both heads.
__global__ void k_heads(const float* __restrict__ pooled,
                        const float* __restrict__ cnt,
                        const float* __restrict__ gfeat,
                        const float* __restrict__ Wo1, const float* __restrict__ bo1,
                        const float* __restrict__ Wo2, const float* __restrict__ bo2,
                        const float* __restrict__ Wb1, const float* __restrict__ bb1,
                        const float* __restrict__ Wb2, const float* __restrict__ bb2,
                        float* __restrict__ out) {
    int g = blockIdx.x;
    int j = threadIdx.x;                                  // 0..31
    float inv = 1.0f / fmaxf(cnt[g], 1.0f);
    float aO = bo1[j], aB = bb1[j];
#pragma unroll 8
    for (int c = 0; c < HID + GFEAT; ++c) {
        float v = (c < HID) ? pooled[(size_t)g * HID + c] * inv
                            : gfeat[(size_t)g * GFEAT + (c - HID)];
        aO += v * Wo1[c * 32 + j];
        aB += v * Wb1[c * 32 + j];
    }
    aO = fmaxf(aO, 0.0f) * Wo2[j];
    aB = fmaxf(aB, 0.0f) * Wb2[j];
#pragma unroll
    for (int off = 16; off > 0; off >>= 1) {
        aO += __shfl_xor(aO, off, 32);
        aB += __shfl_xor(aB, off, 32);
    }
    if (j == 0) {
        out[g]          = aO + bo2[0];                    // orange
        out[NGRAPH + g] = aB + bb2[0];                    // blue
    }
}

// ===========================================================================
extern "C" void kernel_launch(void* const* d_in, const int* in_sizes, int n_in,
                              void* d_out, int out_size, void* d_ws, size_t ws_size,
                              hipStream_t stream) {
    const float* x      = (const float*)d_in[0];
    const int*   ei     = (const int*)  d_in[1];
    const int*   batch  = (const int*)  d_in[2];
    const float* gfeat  = (const float*)d_in[3];
    const float* W1     = (const float*)d_in[4];
    const float* b1     = (const float*)d_in[5];
    const float* gamma1 = (const float*)d_in[6];
    const float* beta1  = (const float*)d_in[7];
    const float* W2     = (const float*)d_in[8];
    const float* b2     = (const float*)d_in[9];
    const float* gamma2 = (const float*)d_in[10];
    const float* beta2  = (const float*)d_in[11];
    const float* Wo1 = (const float*)d_in[12];
    const float* bo1 = (const float*)d_in[13];
    const float* Wo2 = (const float*)d_in[14];
    const float* bo2 = (const float*)d_in[15];
    const float* Wb1 = (const float*)d_in[16];
    const float* bb1 = (const float*)d_in[17];
    const float* Wb2 = (const float*)d_in[18];
    const float* bb2 = (const float*)d_in[19];

    const int* src = ei;              // edge_index[0, :]
    const int* dst = ei + NEDGES;     // edge_index[1, :]
    float* out = (float*)d_out;

    // --- workspace layout (floats) ---
    float* ws     = (float*)d_ws;
    float* dinv   = ws;                              // NNODES
    float* xw     = dinv  + NNODES;                  // NNODES*64
    float* h1     = xw    + (size_t)NNODES * HID;    // NNODES*64
    float* h2     = h1    + (size_t)NNODES * HID;    // NNODES*64
    float* bnsum  = h2    + (size_t)NNODES * HID;    // 64
    float* bnsq   = bnsum + HID;                     // 64  (contiguous w/ bnsum)
    float* pooled = bnsq  + HID;                     // NGRAPH*64
    float* cnt    = pooled + (size_t)NGRAPH * HID;   // NGRAPH (contig w/ pooled)

    const int T = 256;
    const int nNF = NNODES * HID;                    // 8388608
    const int nEF = NEDGES * HID;                    // 134217728 < 2^31
    const int gemmWaves = NNODES / 16;               // 8192 waves, 16x64 strip

    // 1. degree (init 1.0 for self loop) -> dinv_sqrt
    k_fill<<<(NNODES + T - 1) / T, T, 0, stream>>>(dinv, NNODES, 1.0f);
    k_degree<<<(NEDGES + T - 1) / T, T, 0, stream>>>(dst, dinv);
    k_rsqrt_inplace<<<(NNODES + T - 1) / T, T, 0, stream>>>(dinv);

    // ---------------- layer 1 ----------------
    k_gemm_wmma<<<gemmWaves * 32 / T, T, 0, stream>>>(x, W1, xw, NNODES, FIN);
    k_fill<<<(nNF + T - 1) / T, T, 0, stream>>>(h1, nNF, 0.0f);
    k_fill<<<1, 128, 0, stream>>>(bnsum, 2 * HID, 0.0f);
    k_edge_scatter<<<nEF / T, T, 0, stream>>>(xw, src, dst, dinv, h1);
    k_selfloop_bias_stats<<<512, 256, 0, stream>>>(h1, xw, dinv, b1, bnsum, bnsq);
    k_bn_relu<<<nNF / T, T, 0, stream>>>(h1, bnsum, bnsq, gamma1, beta1);

    // ---------------- layer 2 ----------------
    k_gemm_wmma<<<gemmWaves * 32 / T, T, 0, stream>>>(h1, W2, xw, NNODES, HID);
    k_fill<<<(nNF + T - 1) / T, T, 0, stream>>>(h2, nNF, 0.0f);
    k_fill<<<1, 128, 0, stream>>>(bnsum, 2 * HID, 0.0f);
    k_edge_scatter<<<nEF / T, T, 0, stream>>>(xw, src, dst, dinv, h2);
    k_selfloop_bias_stats<<<512, 256, 0, stream>>>(h2, xw, dinv, b2, bnsum, bnsq);
    k_bn_relu<<<nNF / T, T, 0, stream>>>(h2, bnsum, bnsq, gamma2, beta2);

    // ---------------- pool + heads ----------------
    k_fill<<<(NGRAPH * HID + NGRAPH + T - 1) / T, T, 0, stream>>>(
        pooled, NGRAPH * HID + NGRAPH, 0.0f);
    k_pool_sum<<<nNF / T, T, 0, stream>>>(h2, batch, pooled);
    k_pool_cnt<<<(NNODES + T - 1) / T, T, 0, stream>>>(batch, cnt);
    k_heads<<<NGRAPH, 32, 0, stream>>>(pooled, cnt, gfeat,
                                       Wo1, bo1, Wo2, bo2,
                                       Wb1, bb1, Wb2, bb2, out);
    (void)in_sizes; (void)n_in; (void)out_size; (void)ws_size;
}